// RandLANet_5265629905071
// MI455X (gfx1250) — compile-verified
//
#include <hip/hip_runtime.h>
#include <math.h>

// ---------------------------------------------------------------------------
// RandLANet attentive pooling block for MI455X (gfx1250, wave32).
//
// Roofline: ~3.4 GFLOP total, ~4.5 MB live data -> L2-resident (192 MB) at
// 23.3 TB/s; compute mapped onto V_WMMA_F32_16X16X4_F32 (exact f32 so KNN
// ordering and softmax match the f32 reference):
//   * KNN gram matrix: inner dim 3 padded to 4 -> one wmma per 16x16 tile,
//     two tiles (32 queries) per wave so all 32 lanes own a selection row.
//   * attention scores: [80 x 64] @ [64 x 64]^T -> 16 chained X4 wmmas/tile.
// Top-K selection is an LDS-resident insertion sort with a register-cached
// worst element: expected insertions/row ~ K*ln(N/K) ~ 106 of 4096, so the
// dynamic shift loop is cheap and keeps static code + VGPR pressure tiny
// (round 1's register-resident unrolled insert exploded into ~17k VALU ops).
// Workspace layout (needs ~3.41 MB of d_ws):
//   [0)                 int   neigh[B*N*K]        (1,310,720 B)
//   [1,310,720)         float pre  [B*OUTC*N]     (2,097,152 B)
//   [3,407,872)         float stats[2*OUTC]       (      256 B)
// ---------------------------------------------------------------------------

#define Bn   4
#define Nn   4096
#define KNN  20
#define INC  32
#define OUTC 32
#define C2   64          // 2*INC
#define EPSV 1e-5f
#define QPB  4           // queries per block in the attention kernel (4*20=80 rows)
#define QKN  32          // queries per wave in the KNN kernel (2 x 16-row tiles)

typedef float v2f __attribute__((ext_vector_type(2)));
typedef float v8f __attribute__((ext_vector_type(8)));

static __device__ __forceinline__ v8f wmma_f32_16x16x4(v2f a, v2f b, v8f c) {
  // (neg_a, A, neg_b, B, c_mod, C, reuse_a, reuse_b)
  return __builtin_amdgcn_wmma_f32_16x16x4_f32(false, a, false, b, (short)0, c,
                                               false, false);
}

// ---------------------------------------------------------------------------
// Kernel 1: brute-force KNN. One wave handles 32 query points of one batch
// (two 16x4 f32 A-tiles). A layout: lanes 0-15 carry K=0(x),K=1(y); lanes
// 16-31 carry K=2(z),K=3(0). Per column tile: two WMMAs -> 32x16 key tile in
// LDS; key = |pn|^2 - 2*pm.pn (row-constant |pm|^2 dropped: ordering is
// unchanged and self stays the strict minimum, matching the reference where
// self is the nearest neighbor). Each lane then insertion-sorts its row's
// 16 candidates into an LDS-resident sorted top-20 list.
// ---------------------------------------------------------------------------
__global__ void __launch_bounds__(32) knn_kernel(const float* __restrict__ x,
                                                 int* __restrict__ neigh) {
  const int lane = threadIdx.x;
  const int tile = blockIdx.x;
  const int b    = tile / (Nn / QKN);
  const int q0   = (tile % (Nn / QKN)) * QKN;
  const float* xb = x + (size_t)b * 3 * Nn;

  // Two A-tiles: queries q0..q0+15 and q0+16..q0+31.
  const int m0 = q0 + (lane & 15);
  v2f a0, a1;
  if (lane < 16) {
    a0.x = xb[m0];          a0.y = xb[Nn + m0];
    a1.x = xb[m0 + 16];     a1.y = xb[Nn + m0 + 16];
  } else {
    a0.x = xb[2 * Nn + m0];      a0.y = 0.0f;
    a1.x = xb[2 * Nn + m0 + 16]; a1.y = 0.0f;
  }

  __shared__ float keytile[QKN][17];    // padded vs bank conflicts
  __shared__ float sBest[QKN][KNN];     // sorted ascending per row
  __shared__ int   sBidx[QKN][KNN];

#pragma unroll
  for (int k = 0; k < KNN; ++k) { sBest[lane][k] = 3.0e38f; sBidx[lane][k] = 0; }
  float worst = 3.0e38f;                // register cache of sBest[lane][KNN-1]
  __syncthreads();

  for (int nt = 0; nt < Nn / 16; ++nt) {
    const int c0 = nt * 16;
    const int n  = c0 + (lane & 15);
    const float px = xb[n], py = xb[Nn + n], pz = xb[2 * Nn + n];
    v2f bb;
    if (lane < 16) { bb.x = px; bb.y = py; }
    else           { bb.x = pz; bb.y = 0.0f; }
    const float sqn = px * px + py * py + pz * pz;

    v8f c0acc = {};
    v8f c1acc = {};
    c0acc = wmma_f32_16x16x4(a0, bb, c0acc);  // rows 0..15 of this wave
    c1acc = wmma_f32_16x16x4(a1, bb, c1acc);  // rows 16..31

#pragma unroll
    for (int v = 0; v < 8; ++v) {
      const int mm = v + ((lane >> 4) << 3);
      keytile[mm][lane & 15]      = sqn - 2.0f * c0acc[v];
      keytile[mm + 16][lane & 15] = sqn - 2.0f * c1acc[v];
    }
    __syncthreads();

    // Each lane owns row `lane`: scan 16 candidates, LDS insertion sort.
    for (int j = 0; j < 16; ++j) {
      const float d = keytile[lane][j];
      if (d < worst) {
        int t = KNN - 1;
        while (t > 0 && sBest[lane][t - 1] > d) {
          sBest[lane][t] = sBest[lane][t - 1];
          sBidx[lane][t] = sBidx[lane][t - 1];
          --t;
        }
        sBest[lane][t] = d;
        sBidx[lane][t] = c0 + j;
        worst = sBest[lane][KNN - 1];
      }
    }
    __syncthreads();
  }

  const int q = q0 + lane;
#pragma unroll
  for (int k = 0; k < KNN; ++k)
    neigh[((size_t)b * Nn + q) * KNN + k] = sBidx[lane][k];
}

// ---------------------------------------------------------------------------
// Kernel 0: zero the batch-norm accumulators (graph replays must be pure).
// ---------------------------------------------------------------------------
__global__ void zero_stats_kernel(float* __restrict__ stats) {
  if (threadIdx.x < 2 * OUTC) stats[threadIdx.x] = 0.0f;
}

// ---------------------------------------------------------------------------
// Kernel 2: per-point attention + projections. One wave per QPB=4 queries
// (80 rows = 5 exact 16-row WMMA M-tiles). LDS ~61 KB/block.
// ---------------------------------------------------------------------------
__global__ void __launch_bounds__(32) attn_kernel(
    const float* __restrict__ x, const float* __restrict__ feature,
    const int* __restrict__ neigh, const float* __restrict__ w_mlp,
    const float* __restrict__ b_mlp, const float* __restrict__ w_att,
    const float* __restrict__ w_conv, const float* __restrict__ b_conv,
    const float* __restrict__ w_out, const float* __restrict__ b_out,
    float* __restrict__ pre, float* __restrict__ stats) {
  const int lane = threadIdx.x;
  const int blk  = blockIdx.x;
  const int b    = blk / (Nn / QPB);
  const int q0   = (blk % (Nn / QPB)) * QPB;
  const float* xb = x + (size_t)b * 3 * Nn;
  const float* fb = feature + (size_t)b * INC * Nn;   // fb[c*Nn + n]
  const int*   nb = neigh + ((size_t)b * Nn + q0) * KNN;

  __shared__ float sW[C2 * C2];              // staged w_att         (16 KB)
  __shared__ float sFeat[QPB * KNN][C2];     // feats rows           (20 KB)
  __shared__ float sScore[QPB * KNN][C2];    // scores -> attn       (20 KB)
  __shared__ float sXF[QPB * KNN][8];        // 7 geometry feats     (2.5 KB)
  __shared__ int   sNI[QPB * KNN];
  __shared__ float sPooled[QPB][C2];
  __shared__ float sStats[2 * OUTC];

  __builtin_prefetch(w_conv, 0, 0);          // global_prefetch_b8
  __builtin_prefetch(w_out, 0, 0);

  for (int i = lane; i < C2 * C2; i += 32) sW[i] = w_att[i];

  // Phase A: geometry per (query, neighbor) row. Center = neighbor 0
  // (= self, since self distance is minimal), matching x_repeat.
  for (int r = lane; r < QPB * KNN; r += 32) {
    const int q    = r / KNN;
    const int nidx = nb[q * KNN + (r % KNN)];
    const int n0   = nb[q * KNN];
    const float cx = xb[n0], cy = xb[Nn + n0], cz = xb[2 * Nn + n0];
    const float rx = xb[nidx] - cx;
    const float ry = xb[Nn + nidx] - cy;
    const float rz = xb[2 * Nn + nidx] - cz;
    const float dn = sqrtf(rx * rx + ry * ry + rz * rz);
    sXF[r][0] = cx; sXF[r][1] = cy; sXF[r][2] = cz;
    sXF[r][3] = rx; sXF[r][4] = ry; sXF[r][5] = rz;
    sXF[r][6] = dn; sXF[r][7] = 0.0f;
    sNI[r] = nidx;
  }
  __syncthreads();

  // Phase B: feats = [gathered features (0..31) | encoded geometry (32..63)]
  for (int i = lane; i < QPB * KNN * C2; i += 32) {
    const int r  = i >> 6;
    const int cc = i & 63;
    float v;
    if (cc < INC) {
      v = fb[(size_t)cc * Nn + sNI[r]];
    } else {
      const int o = cc - INC;
      v = b_mlp[o];
#pragma unroll
      for (int f = 0; f < 7; ++f) v += w_mlp[o * 7 + f] * sXF[r][f];
    }
    sFeat[r][cc] = v;
  }
  __syncthreads();

  // Phase C: scores[r][d] = sum_c feats[r][c] * w_att[d][c] via f32 WMMA.
  // 5 M-tiles x 4 N-tiles x 16 K-steps = 320 wmma per wave.
  for (int mt = 0; mt < (QPB * KNN) / 16; ++mt) {
    for (int dt = 0; dt < C2 / 16; ++dt) {
      v8f acc = {};
#pragma unroll
      for (int kk = 0; kk < C2 / 4; ++kk) {
        const int k0   = kk * 4 + ((lane >> 4) << 1);  // K=0/1 or 2/3 per half-wave
        const int arow = mt * 16 + (lane & 15);
        const int dcol = dt * 16 + (lane & 15);
        v2f av, bv;
        av.x = sFeat[arow][k0];     av.y = sFeat[arow][k0 + 1];
        bv.x = sW[dcol * C2 + k0];  bv.y = sW[dcol * C2 + k0 + 1];
        acc = wmma_f32_16x16x4(av, bv, acc);
      }
#pragma unroll
      for (int v = 0; v < 8; ++v)
        sScore[mt * 16 + v + ((lane >> 4) << 3)][dt * 16 + (lane & 15)] = acc[v];
    }
  }
  __syncthreads();

  // Phase D: softmax over the 64 channels (torch dim=1), per row.
  for (int r = lane; r < QPB * KNN; r += 32) {
    float mx = -3.0e38f;
    for (int cc = 0; cc < C2; ++cc) mx = fmaxf(mx, sScore[r][cc]);
    float s = 0.0f;
    for (int cc = 0; cc < C2; ++cc) {
      const float e = __expf(sScore[r][cc] - mx);
      sScore[r][cc] = e;
      s += e;
    }
    const float inv = 1.0f / s;
    for (int cc = 0; cc < C2; ++cc) sScore[r][cc] *= inv;
  }
  __syncthreads();

  // Phase E: pooled[q][c] = sum_k attn * feats
  for (int i = lane; i < QPB * C2; i += 32) {
    const int q = i >> 6, cc = i & 63;
    float s = 0.0f;
#pragma unroll
    for (int k = 0; k < KNN; ++k)
      s += sScore[q * KNN + k][cc] * sFeat[q * KNN + k][cc];
    sPooled[q][cc] = s;
  }
  if (lane < 2 * OUTC) sStats[lane] = 0.0f;
  __syncthreads();

  // Phase F: out = pooled @ w_conv^T + b_conv + w_out @ feature + b_out
  for (int i = lane; i < QPB * OUTC; i += 32) {
    const int q = i >> 5, o = i & 31;
    float s = b_conv[o] + b_out[o];
#pragma unroll
    for (int cc = 0; cc < C2; ++cc) s += sPooled[q][cc] * w_conv[o * C2 + cc];
#pragma unroll
    for (int cc = 0; cc < INC; ++cc) s += w_out[o * INC + cc] * fb[(size_t)cc * Nn + (q0 + q)];
    pre[((size_t)b * OUTC + o) * Nn + (q0 + q)] = s;
    atomicAdd(&sStats[o], s);
    atomicAdd(&sStats[OUTC + o], s * s);
  }
  __syncthreads();
  if (lane < 2 * OUTC) atomicAdd(&stats[lane], sStats[lane]);
}

// ---------------------------------------------------------------------------
// Kernel 3: apply training-mode BatchNorm1d with the accumulated batch stats.
// ---------------------------------------------------------------------------
__global__ void bn_kernel(const float* __restrict__ pre,
                          const float* __restrict__ stats,
                          const float* __restrict__ gamma,
                          const float* __restrict__ beta,
                          float* __restrict__ out) {
  const int i = blockIdx.x * blockDim.x + threadIdx.x;
  const int total = Bn * OUTC * Nn;
  if (i < total) {
    const int o = (i / Nn) % OUTC;
    const float cnt  = (float)(Bn * Nn);
    const float mean = stats[o] / cnt;
    const float var  = stats[OUTC + o] / cnt - mean * mean;
    out[i] = (pre[i] - mean) * rsqrtf(var + EPSV) * gamma[o] + beta[o];
  }
}

// ---------------------------------------------------------------------------
extern "C" void kernel_launch(void* const* d_in, const int* in_sizes, int n_in,
                              void* d_out, int out_size, void* d_ws, size_t ws_size,
                              hipStream_t stream) {
  const float* x       = (const float*)d_in[0];
  const float* feature = (const float*)d_in[1];
  const float* w_mlp   = (const float*)d_in[2];
  const float* b_mlp   = (const float*)d_in[3];
  const float* w_att   = (const float*)d_in[4];
  const float* w_conv  = (const float*)d_in[5];
  const float* b_conv  = (const float*)d_in[6];
  const float* w_out   = (const float*)d_in[7];
  const float* b_out   = (const float*)d_in[8];
  const float* gamma   = (const float*)d_in[9];
  const float* beta    = (const float*)d_in[10];
  float* out = (float*)d_out;

  int*   neigh = (int*)d_ws;
  float* pre   = (float*)((char*)d_ws + (size_t)Bn * Nn * KNN * sizeof(int));
  float* stats = pre + (size_t)Bn * OUTC * Nn;

  zero_stats_kernel<<<1, 2 * OUTC, 0, stream>>>(stats);
  knn_kernel<<<Bn * (Nn / QKN), 32, 0, stream>>>(x, neigh);
  attn_kernel<<<Bn * (Nn / QPB), 32, 0, stream>>>(
      x, feature, neigh, w_mlp, b_mlp, w_att, w_conv, b_conv, w_out, b_out,
      pre, stats);
  const int total = Bn * OUTC * Nn;
  bn_kernel<<<(total + 255) / 256, 256, 0, stream>>>(pre, stats, gamma, beta, out);
}